// VolumeLoss_46119358824512
// MI455X (gfx1250) — compile-verified
//
#include <hip/hip_runtime.h>
#include <hip/hip_bf16.h>
#include <math.h>

#define EPSF     0.0001f
#define SEPF     0.01f
#define NORM_EPS 1e-12f

typedef float v2f __attribute__((ext_vector_type(2)));
typedef float v8f __attribute__((ext_vector_type(8)));

// -------------------------------------------------------------------------
// Prep: build augmented fp32 operands for the K=4 fp32 WMMA:
//   Naug[f]   = (nx, ny, nz, d1)   with d1 = -n·verts[v0]
//   Uka[f]    = (x, y, z, 1)       corner k of face f
// Padded rows (f >= F) are zero so WMMA produces DU=0 -> keep=false.
// Also zeroes the row counters (required: harness poisons ws once only).
// -------------------------------------------------------------------------
__global__ void vl_prep(const float* __restrict__ verts,
                        const float* __restrict__ normals,
                        const int*   __restrict__ faces,
                        int F, int F16,
                        float4* __restrict__ Naug,
                        float4* __restrict__ U0a,
                        float4* __restrict__ U1a,
                        float4* __restrict__ U2a,
                        unsigned* __restrict__ rowcnt) {
    int f = blockIdx.x * blockDim.x + threadIdx.x;
    if (f >= F16) return;
    rowcnt[f] = 0u;
    if (f < F) {
        int i0 = faces[3 * f + 0];
        int i1 = faces[3 * f + 1];
        int i2 = faces[3 * f + 2];
        float nx = normals[3 * f + 0];
        float ny = normals[3 * f + 1];
        float nz = normals[3 * f + 2];
        float p0x = verts[3 * i0 + 0], p0y = verts[3 * i0 + 1], p0z = verts[3 * i0 + 2];
        float p1x = verts[3 * i1 + 0], p1y = verts[3 * i1 + 1], p1z = verts[3 * i1 + 2];
        float p2x = verts[3 * i2 + 0], p2y = verts[3 * i2 + 1], p2z = verts[3 * i2 + 2];
        float d1 = -(nx * p0x + ny * p0y + nz * p0z);
        Naug[f] = make_float4(nx, ny, nz, d1);
        U0a[f]  = make_float4(p0x, p0y, p0z, 1.0f);
        U1a[f]  = make_float4(p1x, p1y, p1z, 1.0f);
        U2a[f]  = make_float4(p2x, p2y, p2z, 1.0f);
    } else {
        float4 z = make_float4(0.f, 0.f, 0.f, 0.f);
        Naug[f] = z; U0a[f] = z; U1a[f] = z; U2a[f] = z;
    }
}

// Möller interval endpoints (branch replicated with selects, as in reference)
__device__ __forceinline__ void vl_intervals(float vp0, float vp1, float vp2,
                                             float vd0, float vd1, float vd2,
                                             float& lo, float& hi) {
    bool c1 = (vd0 * vd1) > 0.0f;
    bool c2 = (!c1) && ((vd0 * vd2) > 0.0f);
    float a0  = c1 ? vp2 : (c2 ? vp1 : vp0);
    float a1  = (c1 || c2) ? vp0 : vp1;
    float a2  = c1 ? vp1 : vp2;
    float d0  = c1 ? vd2 : (c2 ? vd1 : vd0);
    float d1_ = (c1 || c2) ? vd0 : vd1;
    float d2  = c1 ? vd1 : vd2;
    float i0 = a0 + (a1 - a0) * (d0 / (d0 - d1_ + EPSF));
    float i1 = a0 + (a2 - a0) * (d0 / (d0 - d2 + EPSF));
    lo = fminf(i0, i1);
    hi = fmaxf(i0, i1);
}

// -------------------------------------------------------------------------
// One wave (32 lanes) = one 16x16 tile of face pairs. Only tj >= ti tiles run
// (final keep matrix is symmetric). 6x V_WMMA_F32_16X16X4_F32 compute the
// plane-distance matrices DU[i,j,k] and DU[j,i,k]; the rest is VALU.
//
// fp32 16x16x4 operand layout (ISA 7.12.2):
//   A (16x4): lane L holds row M = L&15; v[0] = K=(L<16?0:2), v[1] = K=(L<16?1:3)
//   B (4x16): mirrored: lane L holds col N = L&15; v[0]=K0|K2, v[1]=K1|K3
//   C/D      : lane L holds col N = L&15, v[r] = row M = r + 8*(L>=16)
// -------------------------------------------------------------------------
__global__ __launch_bounds__(32)
void vl_pairs(const float4* __restrict__ Naug,
              const float4* __restrict__ U0a,
              const float4* __restrict__ U1a,
              const float4* __restrict__ U2a,
              unsigned* __restrict__ rowcnt,
              int F, int ntiles) {
    int tile = blockIdx.x;
    int ti = tile / ntiles;
    int tj = tile % ntiles;
    if (tj < ti) return;                  // scalar branch: EXEC stays all-ones

    const int lane = threadIdx.x;         // 0..31 (wave32)
    const int half = lane >> 4;           // 0: rows 0-7 / K0,K1 ; 1: rows 8-15 / K2,K3
    const int l15  = lane & 15;
    const int i    = ti * 16 + l15;       // my row index (for operand build / LDS)
    const int j    = tj * 16 + l15;       // my column index (pair work)

    float4 Ni  = Naug[i], Nj  = Naug[j];
    float4 U0i = U0a[i],  U1i = U1a[i],  U2i = U2a[i];
    float4 U0j = U0a[j],  U1j = U1a[j],  U2j = U2a[j];

    // WMMA operands
    v2f aN, aU0, aU1, aU2, bN, bU0, bU1, bU2;
    aN[0]  = half ? Ni.z  : Ni.x;   aN[1]  = half ? Ni.w  : Ni.y;
    aU0[0] = half ? U0i.z : U0i.x;  aU0[1] = half ? U0i.w : U0i.y;
    aU1[0] = half ? U1i.z : U1i.x;  aU1[1] = half ? U1i.w : U1i.y;
    aU2[0] = half ? U2i.z : U2i.x;  aU2[1] = half ? U2i.w : U2i.y;
    bN[0]  = half ? Nj.z  : Nj.x;   bN[1]  = half ? Nj.w  : Nj.y;
    bU0[0] = half ? U0j.z : U0j.x;  bU0[1] = half ? U0j.w : U0j.y;
    bU1[0] = half ? U1j.z : U1j.x;  bU1[1] = half ? U1j.w : U1j.y;
    bU2[0] = half ? U2j.z : U2j.x;  bU2[1] = half ? U2j.w : U2j.y;

    v8f zc = {};
    // DUij_k[m][n] = Naug[i_base+m] . Ukaug[j_base+n]  = DU[i, j, k]
    v8f DUij0 = __builtin_amdgcn_wmma_f32_16x16x4_f32(false, aN, false, bU0, (short)0, zc, false, false);
    v8f DUij1 = __builtin_amdgcn_wmma_f32_16x16x4_f32(false, aN, false, bU1, (short)0, zc, false, false);
    v8f DUij2 = __builtin_amdgcn_wmma_f32_16x16x4_f32(false, aN, false, bU2, (short)0, zc, false, false);
    // DUji_k[m][n] = Ukaug[i_base+m] . Naug[j_base+n]  = DU[j, i, k]
    v8f DUji0 = __builtin_amdgcn_wmma_f32_16x16x4_f32(false, aU0, false, bN, (short)0, zc, false, false);
    v8f DUji1 = __builtin_amdgcn_wmma_f32_16x16x4_f32(false, aU1, false, bN, (short)0, zc, false, false);
    v8f DUji2 = __builtin_amdgcn_wmma_f32_16x16x4_f32(false, aU2, false, bN, (short)0, zc, false, false);

    // Stage row-side raw data (n_i and the 3 corners of triangle i) in LDS
    __shared__ float rowd[16][12];
    if (lane < 16) {
        rowd[lane][0]  = Ni.x;  rowd[lane][1]  = Ni.y;  rowd[lane][2]  = Ni.z;
        rowd[lane][3]  = U0i.x; rowd[lane][4]  = U0i.y; rowd[lane][5]  = U0i.z;
        rowd[lane][6]  = U1i.x; rowd[lane][7]  = U1i.y; rowd[lane][8]  = U1i.z;
        rowd[lane][9]  = U2i.x; rowd[lane][10] = U2i.y; rowd[lane][11] = U2i.z;
    }
    __syncthreads();

    const float njx = Nj.x, njy = Nj.y, njz = Nj.z;
    const bool  jvalid = (j < F);
    unsigned colsum = 0;

#pragma unroll
    for (int r = 0; r < 8; ++r) {
        const int   mi = r + half * 8;
        const int   ii = ti * 16 + mi;
        const float nix = rowd[mi][0], niy = rowd[mi][1], niz = rowd[mi][2];

        // D = cross(n_i, n_j), normalized with max(||D||, 1e-12)
        float cx = niy * njz - niz * njy;
        float cy = niz * njx - nix * njz;
        float cz = nix * njy - niy * njx;
        float inv = 1.0f / fmaxf(sqrtf(cx * cx + cy * cy + cz * cz), NORM_EPS);
        float dx = cx * inv, dy = cy * inv, dz = cz * inv;

        float duij0 = DUij0[r], duij1 = DUij1[r], duij2 = DUij2[r];
        float duji0 = DUji0[r], duji1 = DUji1[r], duji2 = DUji2[r];

        // projections of triangle j's corners and triangle i's corners onto D
        float upj0 = dx * U0j.x + dy * U0j.y + dz * U0j.z;
        float upj1 = dx * U1j.x + dy * U1j.y + dz * U1j.z;
        float upj2 = dx * U2j.x + dy * U2j.y + dz * U2j.z;
        float upi0 = dx * rowd[mi][3] + dy * rowd[mi][4]  + dz * rowd[mi][5];
        float upi1 = dx * rowd[mi][6] + dy * rowd[mi][7]  + dz * rowd[mi][8];
        float upi2 = dx * rowd[mi][9] + dy * rowd[mi][10] + dz * rowd[mi][11];

        float lo0, hi0, lo1, hi1;
        vl_intervals(upj0, upj1, upj2, duij0, duij1, duij2, lo0, hi0);
        vl_intervals(upi0, upi1, upi2, duji0, duji1, duji2, lo1, hi1);

        bool condij = !(((duij0 > -EPSF) && (duij1 > -EPSF) && (duij2 > -EPSF)) ||
                        ((duij0 <  EPSF) && (duij1 <  EPSF) && (duij2 <  EPSF)));
        bool condji = !(((duji0 > -EPSF) && (duji1 > -EPSF) && (duji2 > -EPSF)) ||
                        ((duji0 <  EPSF) && (duji1 <  EPSF) && (duji2 <  EPSF)));
        bool keep0 = condij && condji;
        bool sepb  = (hi0 <= lo1 + SEPF) || (hi1 <= lo0 + SEPF);
        bool keepf = (ii < F) && jvalid && keep0 && ((ii == j) || !sepb);

        unsigned v = keepf ? 1u : 0u;
        colsum += v;
        // reduce over the 16 lanes of each half (same row ii, 16 columns)
        v += __shfl_xor(v, 1, 32);
        v += __shfl_xor(v, 2, 32);
        v += __shfl_xor(v, 4, 32);
        v += __shfl_xor(v, 8, 32);
        if (l15 == 0 && ii < F) atomicAdd(&rowcnt[ii], v);
    }

    if (ti != tj) {
        // symmetric contribution: column sums of this tile feed row[j]
        colsum += __shfl_xor(colsum, 16, 32);
        if (half == 0 && jvalid) atomicAdd(&rowcnt[j], colsum);
    }
}

// -------------------------------------------------------------------------
// loss = ( sum_f x[v0[f]] * row[f]  +  2 * sum_f x[v2[f]] * row[f] ) / 3
// -------------------------------------------------------------------------
__global__ void vl_loss(const float* __restrict__ verts,
                        const int*   __restrict__ faces,
                        const unsigned* __restrict__ rowcnt,
                        int F, float* __restrict__ out) {
    __shared__ float sbuf[256];
    float acc = 0.0f;
    for (int f = threadIdx.x; f < F; f += 256) {
        float r = (float)rowcnt[f];
        int v0 = faces[3 * f + 0];
        int v2 = faces[3 * f + 2];
        acc += (verts[3 * v0 + 0] + 2.0f * verts[3 * v2 + 0]) * r;
    }
    sbuf[threadIdx.x] = acc;
    __syncthreads();
    for (int s = 128; s > 0; s >>= 1) {
        if (threadIdx.x < s) sbuf[threadIdx.x] += sbuf[threadIdx.x + s];
        __syncthreads();
    }
    if (threadIdx.x == 0) out[0] = sbuf[0] / 3.0f;
}

extern "C" void kernel_launch(void* const* d_in, const int* in_sizes, int n_in,
                              void* d_out, int out_size, void* d_ws, size_t ws_size,
                              hipStream_t stream) {
    const float* verts   = (const float*)d_in[0];   // [1, V, 3] f32
    const float* normals = (const float*)d_in[1];   // [1, F, 3] f32
    const int*   faces   = (const int*)d_in[2];     // [F, 3] int

    const int F      = in_sizes[2] / 3;             // 2500
    const int ntiles = (F + 15) / 16;               // 157
    const int F16    = ntiles * 16;                 // 2512

    char* ws = (char*)d_ws;
    float4*   Naug   = (float4*)ws;
    float4*   U0a    = Naug + F16;
    float4*   U1a    = U0a + F16;
    float4*   U2a    = U1a + F16;
    unsigned* rowcnt = (unsigned*)(U2a + F16);      // ~171 KB total

    vl_prep<<<(F16 + 255) / 256, 256, 0, stream>>>(verts, normals, faces, F, F16,
                                                   Naug, U0a, U1a, U2a, rowcnt);
    vl_pairs<<<ntiles * ntiles, 32, 0, stream>>>(Naug, U0a, U1a, U2a, rowcnt, F, ntiles);
    vl_loss<<<1, 256, 0, stream>>>(verts, faces, rowcnt, F, (float*)d_out);
}